// GCN_83210696393488
// MI455X (gfx1250) — compile-verified
//
#include <hip/hip_runtime.h>
#include <math.h>

typedef __attribute__((ext_vector_type(2))) float v2f;
typedef __attribute__((ext_vector_type(8))) float v8f;

#define N_NODES  65536
#define N_EDGES  1048576
#define NUM_PTS  1024
#define N_SHAPES 64
#define BN_EPS   1e-5f

// ---------------------------------------------------------------- utilities
__global__ void fill_zero(float* __restrict__ p, int n) {
    int i = blockIdx.x * blockDim.x + threadIdx.x;
    if (i < n) p[i] = 0.0f;
}

// ------------------------------------------------ edge aggregation (adj @ x)
// agg[dst] += x[src] * val   -- done in the NARROW feature dim (associativity
// rewrite (adj@x)@W), so gc3's aggregation is 64-wide, not 1024-wide.
__global__ void edge_agg3(const float* __restrict__ x, const int* __restrict__ src,
                          const int* __restrict__ dst, const float* __restrict__ val,
                          float* __restrict__ agg) {
    int e = blockIdx.x * blockDim.x + threadIdx.x;
    if (e >= N_EDGES) return;
    int s = src[e], d = dst[e];
    float v = val[e];
    atomicAdd(&agg[d * 3 + 0], x[s * 3 + 0] * v);
    atomicAdd(&agg[d * 3 + 1], x[s * 3 + 1] * v);
    atomicAdd(&agg[d * 3 + 2], x[s * 3 + 2] * v);
}

// 16 lanes per edge, float4 gather per lane (L2-resident: h is only 16MB)
__global__ void edge_agg64(const float* __restrict__ h, const int* __restrict__ src,
                           const int* __restrict__ dst, const float* __restrict__ val,
                           float* __restrict__ agg) {
    int gid = blockIdx.x * blockDim.x + threadIdx.x;
    int e = gid >> 4;
    if (e >= N_EDGES) return;
    int f = (gid & 15) * 4;
    int s = src[e], d = dst[e];
    float v = val[e];
    float4 m = *(const float4*)&h[s * 64 + f];
    atomicAdd(&agg[d * 64 + f + 0], m.x * v);
    atomicAdd(&agg[d * 64 + f + 1], m.y * v);
    atomicAdd(&agg[d * 64 + f + 2], m.z * v);
    atomicAdd(&agg[d * 64 + f + 3], m.w * v);
}

// ------------------------------------------------ gc1: K=3 GEMM + bias + relu
__global__ void gc1_kernel(const float* __restrict__ agg, const float* __restrict__ W,
                           const float* __restrict__ b, float* __restrict__ out) {
    int i = blockIdx.x * blockDim.x + threadIdx.x;
    if (i >= N_NODES * 64) return;
    int n = i >> 6, f = i & 63;
    float r = fmaf(agg[n * 3 + 0], W[0 * 64 + f],
             fmaf(agg[n * 3 + 1], W[1 * 64 + f],
             fmaf(agg[n * 3 + 2], W[2 * 64 + f], b[f])));
    out[i] = fmaxf(r, 0.0f);
}

// ------------------------------------------------ generic fp32 WMMA GEMM
// One wave per 16x16 tile of C = A[M,K] @ B[K,N] + bias, optional relu.
// Uses V_WMMA_F32_16X16X4_F32 (full fp32 matrix path, K stepped by 4).
template <bool RELU>
__global__ void gemm_wmma(const float* __restrict__ A, const float* __restrict__ B,
                          const float* __restrict__ bias, float* __restrict__ C,
                          int M, int N, int K) {
    const int lane = threadIdx.x;        // 0..31
    const int half = lane >> 4;          // 0 or 1
    const int l16  = lane & 15;
    const int col0 = blockIdx.x * 16;
    const int row0 = blockIdx.y * 16;
    (void)M;

    v8f c = {};
    const float* arow = A + (long)(row0 + l16) * K + 2 * half;   // row l16, K-off 2*half
    const float* bcol = B + (long)(2 * half) * N + col0 + l16;   // rows 2*half,2*half+1
    for (int k = 0; k < K; k += 4) {
        float2 at = *(const float2*)(arow + k);                  // A[m][k+2h], A[m][k+2h+1]
        v2f a;  a.x = at.x;          a.y = at.y;
        v2f bv; bv.x = bcol[(long)k * N]; bv.y = bcol[(long)k * N + N];
        c = __builtin_amdgcn_wmma_f32_16x16x4_f32(false, a, false, bv,
                                                  (short)0, c, false, false);
    }
    float bb = bias[col0 + l16];
#pragma unroll
    for (int r = 0; r < 8; r++) {
        float v = c[r] + bb;
        if (RELU) v = fmaxf(v, 0.0f);
        C[(long)(row0 + r + 8 * half) * N + col0 + l16] = v;
    }
}

// -------------------- gc3 fused: pooled[b][n] = relu(max_m (A@W3)[m][n] + b3[n])
// One wave per (shape, 16-feature strip). B fragments (K=64) preloaded in regs.
// Never materializes the 256MB [65536,1024] activation.
__global__ void gc3_pool(const float* __restrict__ A, const float* __restrict__ W,
                         const float* __restrict__ bias, float* __restrict__ pooled) {
    const int lane = threadIdx.x;
    const int half = lane >> 4;
    const int l16  = lane & 15;
    const int col0 = blockIdx.x * 16;
    const int shape = blockIdx.y;

    v2f bfrag[16];
#pragma unroll
    for (int s = 0; s < 16; s++) {
        int k = 4 * s + 2 * half;
        bfrag[s].x = W[(long)k * 1024 + col0 + l16];
        bfrag[s].y = W[(long)(k + 1) * 1024 + col0 + l16];
    }

    float m[8];
#pragma unroll
    for (int r = 0; r < 8; r++) m[r] = -3.4e38f;

    const int base = shape * NUM_PTS;
    for (int mt = 0; mt < NUM_PTS / 16; mt++) {
        v8f c = {};
        const float* arow = A + (long)(base + mt * 16 + l16) * 64 + 2 * half;
#pragma unroll
        for (int s = 0; s < 16; s++) {
            float2 at = *(const float2*)(arow + 4 * s);
            v2f a; a.x = at.x; a.y = at.y;
            c = __builtin_amdgcn_wmma_f32_16x16x4_f32(false, a, false, bfrag[s],
                                                      (short)0, c, false, false);
        }
#pragma unroll
        for (int r = 0; r < 8; r++) m[r] = fmaxf(m[r], c[r]);
    }
    float t = m[0];
#pragma unroll
    for (int r = 1; r < 8; r++) t = fmaxf(t, m[r]);
    t = fmaxf(t, __shfl_xor(t, 16, 32));             // combine M halves (same N)
    float outv = fmaxf(t + bias[col0 + l16], 0.0f);  // bias/relu monotone => after max
    if (lane < 16) pooled[shape * 1024 + col0 + lane] = outv;
}

// ------------------------------------------------ train-mode BN + relu (B=64)
__global__ void bn_relu(const float* __restrict__ X, const float* __restrict__ g,
                        const float* __restrict__ b, float* __restrict__ Y, int F) {
    int c = blockIdx.x * blockDim.x + threadIdx.x;
    if (c >= F) return;
    float s = 0.0f;
    for (int r = 0; r < N_SHAPES; r++) s += X[r * F + c];
    float mean = s / (float)N_SHAPES;
    float vs = 0.0f;
    for (int r = 0; r < N_SHAPES; r++) { float d = X[r * F + c] - mean; vs += d * d; }
    float scale = g[c] * rsqrtf(vs / (float)N_SHAPES + BN_EPS);
    float beta = b[c];
    for (int r = 0; r < N_SHAPES; r++) {
        float v = (X[r * F + c] - mean) * scale + beta;
        Y[r * F + c] = fmaxf(v, 0.0f);
    }
}

// ------------------------------------------------ fc3 (N=40) + log_softmax
__global__ void fc3_kernel(const float* __restrict__ X, const float* __restrict__ W,
                           const float* __restrict__ b, float* __restrict__ S) {
    int i = blockIdx.x * blockDim.x + threadIdx.x;
    if (i >= N_SHAPES * 40) return;
    int r = i / 40, c = i % 40;
    float s = b[c];
    for (int k = 0; k < 256; k++) s = fmaf(X[r * 256 + k], W[k * 40 + c], s);
    S[i] = s;
}

__global__ void logsoftmax_kernel(const float* __restrict__ S, float* __restrict__ out) {
    int r = blockIdx.x * blockDim.x + threadIdx.x;
    if (r >= N_SHAPES) return;
    float mx = -3.4e38f;
    for (int c = 0; c < 40; c++) mx = fmaxf(mx, S[r * 40 + c]);
    float sum = 0.0f;
    for (int c = 0; c < 40; c++) sum += expf(S[r * 40 + c] - mx);
    float lse = mx + logf(sum);
    for (int c = 0; c < 40; c++) out[r * 40 + c] = S[r * 40 + c] - lse;
}

// ---------------------------------------------------------------- launcher
extern "C" void kernel_launch(void* const* d_in, const int* in_sizes, int n_in,
                              void* d_out, int out_size, void* d_ws, size_t ws_size,
                              hipStream_t stream) {
    (void)in_sizes; (void)n_in; (void)out_size; (void)ws_size;
    const float* x     = (const float*)d_in[0];
    const int*   esrc  = (const int*)d_in[1];
    const int*   edst  = (const int*)d_in[2];
    const float* eval_ = (const float*)d_in[3];
    /* d_in[4] = num_pts (fixed 1024 per reference) */
    const float* gc1_w = (const float*)d_in[5];
    const float* gc1_b = (const float*)d_in[6];
    const float* gc2_w = (const float*)d_in[7];
    const float* gc2_b = (const float*)d_in[8];
    const float* gc3_w = (const float*)d_in[9];
    const float* gc3_b = (const float*)d_in[10];
    const float* fc1_w = (const float*)d_in[11];
    const float* fc1_b = (const float*)d_in[12];
    const float* bn1_g = (const float*)d_in[13];
    const float* bn1_b = (const float*)d_in[14];
    const float* fc2_w = (const float*)d_in[15];
    const float* fc2_b = (const float*)d_in[16];
    const float* bn2_g = (const float*)d_in[17];
    const float* bn2_b = (const float*)d_in[18];
    const float* fc3_w = (const float*)d_in[19];
    const float* fc3_b = (const float*)d_in[20];

    // workspace layout (reused buffers; ~35MB total)
    float* bufA   = (float*)d_ws;                 // [65536,64] h1 / h2
    float* bufB   = bufA + (size_t)N_NODES * 64;  // [65536,64] agg1 / agg2
    float* agg0   = bufB + (size_t)N_NODES * 64;  // [65536,3]
    float* pooled = agg0 + (size_t)N_NODES * 3;   // [64,1024]
    float* z1     = pooled + 64 * 1024;           // [64,512]
    float* z1n    = z1 + 64 * 512;
    float* z2     = z1n + 64 * 512;               // [64,256]
    float* z2n    = z2 + 64 * 256;
    float* scores = z2n + 64 * 256;               // [64,40]

    // ---- gc1: agg0 = adj@x ; h1 = relu(agg0 @ W1 + b1)
    fill_zero<<<(N_NODES * 3 + 255) / 256, 256, 0, stream>>>(agg0, N_NODES * 3);
    edge_agg3<<<(N_EDGES + 255) / 256, 256, 0, stream>>>(x, esrc, edst, eval_, agg0);
    gc1_kernel<<<(N_NODES * 64 + 255) / 256, 256, 0, stream>>>(agg0, gc1_w, gc1_b, bufA);

    // ---- gc2: agg1 = adj@h1 ; h2 = relu(agg1 @ W2 + b2)   (WMMA f32)
    fill_zero<<<(N_NODES * 64 + 255) / 256, 256, 0, stream>>>(bufB, N_NODES * 64);
    edge_agg64<<<(N_EDGES * 16 + 255) / 256, 256, 0, stream>>>(bufA, esrc, edst, eval_, bufB);
    gemm_wmma<true><<<dim3(64 / 16, N_NODES / 16), 32, 0, stream>>>(
        bufB, gc2_w, gc2_b, bufA, N_NODES, 64, 64);

    // ---- gc3 + maxpool fused: agg2 = adj@h2 ; pooled = max_pool(relu(agg2@W3+b3))
    fill_zero<<<(N_NODES * 64 + 255) / 256, 256, 0, stream>>>(bufB, N_NODES * 64);
    edge_agg64<<<(N_EDGES * 16 + 255) / 256, 256, 0, stream>>>(bufA, esrc, edst, eval_, bufB);
    gc3_pool<<<dim3(1024 / 16, N_SHAPES), 32, 0, stream>>>(bufB, gc3_w, gc3_b, pooled);

    // ---- MLP head
    gemm_wmma<false><<<dim3(512 / 16, N_SHAPES / 16), 32, 0, stream>>>(
        pooled, fc1_w, fc1_b, z1, N_SHAPES, 512, 1024);
    bn_relu<<<(512 + 255) / 256, 256, 0, stream>>>(z1, bn1_g, bn1_b, z1n, 512);
    gemm_wmma<false><<<dim3(256 / 16, N_SHAPES / 16), 32, 0, stream>>>(
        z1n, fc2_w, fc2_b, z2, N_SHAPES, 256, 512);
    bn_relu<<<1, 256, 0, stream>>>(z2, bn2_g, bn2_b, z2n, 256);
    fc3_kernel<<<(N_SHAPES * 40 + 255) / 256, 256, 0, stream>>>(z2n, fc3_w, fc3_b, scores);
    logsoftmax_kernel<<<1, 64, 0, stream>>>(scores, (float*)d_out);
}